// PrototypicalTransformerNLU_10728828305764
// MI455X (gfx1250) — compile-verified
//
#include <hip/hip_runtime.h>
#include <hip/hip_bf16.h>
#include <math.h>

typedef float v2f __attribute__((ext_vector_type(2)));
typedef float v8f __attribute__((ext_vector_type(8)));

#define B_   128
#define T_   128
#define H_   768
#define E_   300
#define NI_  60
#define NS_  100
#define NTOK (B_*(T_-1))   // 16256

// ---------------- workspace layout (float offsets) ----------------
#define WS_ENC1     0            // 16384*300 = 4,915,200 ; later reused as compacted toks2 (16256*300)
#define WS_ENC2     4915200      // 16384*300
#define WS_SPE      9830400      // slot_proto_enc 100*300
#define WS_ISUM     9860400      // 60*300
#define WS_ICNT     9878400      // 64
#define WS_SSUM     9878464      // 100*300
#define WS_SCNT     9908464      // 128
#define WS_LOSS     9908592      // 16  (0: intent, 1: slot)
#define WS_ZERO_N   48208        // ISUM..LOSS contiguous
#define WS_IPROTO   9908608      // 60*300
#define WS_SPROTO   9926608      // 100*300
#define WS_IPNORM   9956608      // 64
#define WS_SPNORM   9956672      // 128
#define WS_CLSNORM  9956800      // 128
#define WS_TOKNORM  9956928      // 16256

// ---------------- output layout (float offsets) ----------------
#define OUT_ILOGITS 0            // 128*60
#define OUT_SLOGITS 7680         // 16256*100
#define OUT_ILOSS   1633280
#define OUT_SLOSS   1633281
#define OUT_TLOSS   1633282
#define OUT_IPROTO  1633283      // 60*300
#define OUT_ICNT    1651283      // 60
#define OUT_SPROTO  1651343     // 100*300
#define OUT_SCNT    1681343     // 100

__global__ void zero_kernel(float* __restrict__ p, int n) {
    int i = blockIdx.x * blockDim.x + threadIdx.x;
    if (i < n) p[i] = 0.0f;
}

// ------------------------------------------------------------------
// Generic C = A(MxK, lda) * B^T  (B is NxK row-major, ldb) + bias[n]
// One wave per 32x32 C block (2x2 WMMA tiles), fp32 WMMA, K step 4.
// Interior blocks take a select-free fast path (wave-uniform branch);
// edge blocks take a guarded path (VALU selects keep EXEC all-1s).
// ------------------------------------------------------------------
__global__ __launch_bounds__(32) void wmma_gemm_abt(
    const float* __restrict__ A, int lda,
    const float* __restrict__ Bm, int ldb,
    float* __restrict__ C, int ldc,
    int M, int N, int K,
    const float* __restrict__ bias)
{
    const int lane  = threadIdx.x & 31;
    const int n16   = lane & 15;
    const int khalf = (lane >> 4) << 1;          // 0 or 2 (A/B fragment K pair)
    const int tm0 = blockIdx.x * 2;              // first of 2 M tiles
    const int tn0 = blockIdx.y * 2;              // first of 2 N tiles

    const int mrow0 = tm0 * 16 + n16;
    const int mrow1 = mrow0 + 16;
    const int ncol0 = tn0 * 16 + n16;
    const int ncol1 = ncol0 + 16;

    v8f acc00, acc01, acc10, acc11;
    #pragma unroll
    for (int i = 0; i < 8; ++i) { acc00[i] = 0.0f; acc01[i] = 0.0f; acc10[i] = 0.0f; acc11[i] = 0.0f; }

    const bool interior = (tm0 * 16 + 32 <= M) && (tn0 * 16 + 32 <= N);  // wave-uniform

    if (interior) {
        const float* a0 = A  + (size_t)mrow0 * lda + khalf;
        const float* a1 = A  + (size_t)mrow1 * lda + khalf;
        const float* b0 = Bm + (size_t)ncol0 * ldb + khalf;
        const float* b1 = Bm + (size_t)ncol1 * ldb + khalf;
        #pragma unroll 4
        for (int k = 0; k < K; k += 4) {
            v2f av0 = *(const v2f*)(a0 + k);
            v2f av1 = *(const v2f*)(a1 + k);
            v2f bv0 = *(const v2f*)(b0 + k);
            v2f bv1 = *(const v2f*)(b1 + k);
            acc00 = __builtin_amdgcn_wmma_f32_16x16x4_f32(false, av0, false, bv0, (short)0, acc00, false, false);
            acc01 = __builtin_amdgcn_wmma_f32_16x16x4_f32(false, av0, false, bv1, (short)0, acc01, false, false);
            acc10 = __builtin_amdgcn_wmma_f32_16x16x4_f32(false, av1, false, bv0, (short)0, acc10, false, false);
            acc11 = __builtin_amdgcn_wmma_f32_16x16x4_f32(false, av1, false, bv1, (short)0, acc11, false, false);
        }
    } else {
        const bool mv0 = mrow0 < M, mv1 = mrow1 < M;
        const bool nv0 = ncol0 < N, nv1 = ncol1 < N;
        const float* a0 = A  + (size_t)(mv0 ? mrow0 : 0) * lda + khalf;
        const float* a1 = A  + (size_t)(mv1 ? mrow1 : 0) * lda + khalf;
        const float* b0 = Bm + (size_t)(nv0 ? ncol0 : 0) * ldb + khalf;
        const float* b1 = Bm + (size_t)(nv1 ? ncol1 : 0) * ldb + khalf;
        v2f zz; zz.x = 0.0f; zz.y = 0.0f;
        for (int k = 0; k < K; k += 4) {
            v2f av0 = *(const v2f*)(a0 + k);
            v2f av1 = *(const v2f*)(a1 + k);
            v2f bv0 = *(const v2f*)(b0 + k);
            v2f bv1 = *(const v2f*)(b1 + k);
            av0 = mv0 ? av0 : zz;   // VALU select, EXEC stays all-1s for WMMA
            av1 = mv1 ? av1 : zz;
            bv0 = nv0 ? bv0 : zz;
            bv1 = nv1 ? bv1 : zz;
            acc00 = __builtin_amdgcn_wmma_f32_16x16x4_f32(false, av0, false, bv0, (short)0, acc00, false, false);
            acc01 = __builtin_amdgcn_wmma_f32_16x16x4_f32(false, av0, false, bv1, (short)0, acc01, false, false);
            acc10 = __builtin_amdgcn_wmma_f32_16x16x4_f32(false, av1, false, bv0, (short)0, acc10, false, false);
            acc11 = __builtin_amdgcn_wmma_f32_16x16x4_f32(false, av1, false, bv1, (short)0, acc11, false, false);
        }
    }

    // Bias per N subtile (column depends only on lane&15 within each tile)
    float bval0 = 0.0f, bval1 = 0.0f;
    if (bias != nullptr) {
        if (ncol0 < N) bval0 = bias[ncol0];
        if (ncol1 < N) bval1 = bias[ncol1];
    }

    // C/D layout: VGPR r -> row = tile*16 + r + 8*(lane>>4), col = tile*16 + (lane&15)
    const int rb = ((lane >> 4) << 3);
    #pragma unroll
    for (int r = 0; r < 8; ++r) {
        int row0 = tm0 * 16 + rb + r;
        int row1 = row0 + 16;
        if (row0 < M) {
            if (ncol0 < N) C[(size_t)row0 * ldc + ncol0] = acc00[r] + bval0;
            if (ncol1 < N) C[(size_t)row0 * ldc + ncol1] = acc01[r] + bval1;
        }
        if (row1 < M) {
            if (ncol0 < N) C[(size_t)row1 * ldc + ncol0] = acc10[r] + bval0;
            if (ncol1 < N) C[(size_t)row1 * ldc + ncol1] = acc11[r] + bval1;
        }
    }
}

// ---------------- prototype accumulation (segment sums) ----------------
__global__ void accum_intent(const float* __restrict__ enc1,
                             const int* __restrict__ labels,
                             float* __restrict__ isum, float* __restrict__ icnt)
{
    int b = blockIdx.x;
    int lab = labels[b];
    const float* row = enc1 + (size_t)b * T_ * E_;   // token 0 (CLS)
    for (int d = threadIdx.x; d < E_; d += blockDim.x)
        atomicAdd(&isum[lab * E_ + d], row[d]);
    if (threadIdx.x == 0) atomicAdd(&icnt[lab], 1.0f);
}

__global__ void accum_slot(const float* __restrict__ enc1,
                           const int* __restrict__ slab,
                           float* __restrict__ ssum, float* __restrict__ scnt)
{
    int rr = blockIdx.x;                 // 0..16255
    int b = rr / (T_ - 1);
    int t = rr % (T_ - 1) + 1;
    int lab = slab[b * T_ + t];
    const float* row = enc1 + ((size_t)b * T_ + t) * E_;
    for (int d = threadIdx.x; d < E_; d += blockDim.x)
        atomicAdd(&ssum[lab * E_ + d], row[d]);
    if (threadIdx.x == 0) atomicAdd(&scnt[lab], 1.0f);
}

// ---------------- prototype finalize ----------------
__global__ void finalize_intent(const float* __restrict__ iproto_in,
                                const int* __restrict__ icnt_in,
                                const float* __restrict__ isum,
                                const float* __restrict__ icnt,
                                float* __restrict__ out_proto,
                                float* __restrict__ ws_proto,
                                float* __restrict__ out_cnt)
{
    int idx = blockIdx.x * blockDim.x + threadIdx.x;
    if (idx < NI_ * E_) {
        int i = idx / E_;
        float c0 = (float)icnt_in[i];
        float p = (c0 * iproto_in[idx] + isum[idx]) / (c0 + icnt[i]);
        out_proto[idx] = p;
        ws_proto[idx]  = p;
    }
    if (idx < NI_) out_cnt[idx] = (float)E_;   // intent_count = ENCODER_DIM
}

__global__ void finalize_slot(const float* __restrict__ sproto_enc,
                              const int* __restrict__ scnt_in,
                              const float* __restrict__ ssum,
                              const float* __restrict__ scnt,
                              float* __restrict__ out_proto,
                              float* __restrict__ ws_proto,
                              float* __restrict__ out_cnt)
{
    int idx = blockIdx.x * blockDim.x + threadIdx.x;
    if (idx < NS_ * E_) {
        int s = idx / E_;
        float c0 = (float)scnt_in[s];
        float p = (c0 * sproto_enc[idx] + ssum[idx]) / (c0 + scnt[s]);
        out_proto[idx] = p;
        ws_proto[idx]  = p;
    }
    if (idx < NS_) out_cnt[idx] = (float)scnt_in[idx] + scnt[idx];
}

// ---------------- compact enc2 tokens 1..T-1 into dense rows ----------------
__global__ void compact_toks(const float* __restrict__ enc2, float* __restrict__ dst) {
    int idx = blockIdx.x * blockDim.x + threadIdx.x;
    if (idx >= NTOK * E_) return;
    int r = idx / E_, d = idx % E_;
    int b = r / (T_ - 1), t = r % (T_ - 1) + 1;
    dst[idx] = enc2[((size_t)b * T_ + t) * E_ + d];
}

// ---------------- per-row L2 norm (clamped at eps) ----------------
__global__ void row_l2norm(const float* __restrict__ X, int ld, int D,
                           float* __restrict__ out)
{
    int r = blockIdx.x;
    const float* row = X + (size_t)r * ld;
    float s = 0.0f;
    for (int d = threadIdx.x; d < D; d += blockDim.x) { float v = row[d]; s += v * v; }
    __shared__ float sh[128];
    sh[threadIdx.x] = s;
    __syncthreads();
    for (int o = 64; o > 0; o >>= 1) {
        if (threadIdx.x < o) sh[threadIdx.x] += sh[threadIdx.x + o];
        __syncthreads();
    }
    if (threadIdx.x == 0) out[r] = fmaxf(sqrtf(sh[0]), 1e-8f);
}

// ---------------- intent softmax + CE-on-probs (in-place over logits) -------
__global__ void intent_softmax_loss(float* __restrict__ logits,
                                    const float* __restrict__ cn,
                                    const float* __restrict__ pn,
                                    const int* __restrict__ labels,
                                    float* __restrict__ loss_acc)
{
    int b = blockIdx.x, j = threadIdx.x;   // blockDim = 64
    __shared__ float sh[64];
    __shared__ float spick;

    float val = 0.0f, s = -INFINITY;
    if (j < NI_) { val = logits[b * NI_ + j] / (cn[b] * pn[j]); s = val; }

    sh[j] = s; __syncthreads();
    for (int o = 32; o > 0; o >>= 1) { if (j < o) sh[j] = fmaxf(sh[j], sh[j + o]); __syncthreads(); }
    float m = sh[0]; __syncthreads();

    float e = (j < NI_) ? __expf(val - m) : 0.0f;
    sh[j] = e; __syncthreads();
    for (int o = 32; o > 0; o >>= 1) { if (j < o) sh[j] += sh[j + o]; __syncthreads(); }
    float sum = sh[0]; __syncthreads();

    float p = (j < NI_) ? e / sum : 0.0f;
    if (j < NI_) logits[b * NI_ + j] = p;

    // logsumexp over the probabilities themselves
    sh[j] = (j < NI_) ? p : -INFINITY; __syncthreads();
    for (int o = 32; o > 0; o >>= 1) { if (j < o) sh[j] = fmaxf(sh[j], sh[j + o]); __syncthreads(); }
    float m2 = sh[0]; __syncthreads();
    float e2 = (j < NI_) ? __expf(p - m2) : 0.0f;
    sh[j] = e2; __syncthreads();
    for (int o = 32; o > 0; o >>= 1) { if (j < o) sh[j] += sh[j + o]; __syncthreads(); }
    float z = sh[0];
    int lab = labels[b];
    if (j == lab) spick = p;
    __syncthreads();
    if (j == 0) atomicAdd(loss_acc, (m2 + logf(z) - spick) * (1.0f / B_));
}

// ---------------- slot softmax + CE-on-probs (in-place over logits) ---------
__global__ void slot_softmax_loss(float* __restrict__ logits,
                                  const float* __restrict__ tn,
                                  const float* __restrict__ pn,
                                  const int* __restrict__ slab2,
                                  const float* __restrict__ bO,
                                  const float* __restrict__ bX,
                                  float* __restrict__ loss_acc)
{
    int r = blockIdx.x, j = threadIdx.x;   // blockDim = 128
    __shared__ float sh[128];
    __shared__ float spick;
    int b = r / (T_ - 1), t = r % (T_ - 1) + 1;

    float val = 0.0f, s = -INFINITY;
    if (j < NS_) {
        float raw = logits[(size_t)r * NS_ + j];
        if (j == 0)      val = bO[0];
        else if (j == 1) val = bX[0];
        else             val = raw / (tn[r] * pn[j]);
        s = val;
    }

    sh[j] = s; __syncthreads();
    for (int o = 64; o > 0; o >>= 1) { if (j < o) sh[j] = fmaxf(sh[j], sh[j + o]); __syncthreads(); }
    float m = sh[0]; __syncthreads();

    float e = (j < NS_) ? __expf(val - m) : 0.0f;
    sh[j] = e; __syncthreads();
    for (int o = 64; o > 0; o >>= 1) { if (j < o) sh[j] += sh[j + o]; __syncthreads(); }
    float sum = sh[0]; __syncthreads();

    float p = (j < NS_) ? e / sum : 0.0f;
    if (j < NS_) logits[(size_t)r * NS_ + j] = p;

    sh[j] = (j < NS_) ? p : -INFINITY; __syncthreads();
    for (int o = 64; o > 0; o >>= 1) { if (j < o) sh[j] = fmaxf(sh[j], sh[j + o]); __syncthreads(); }
    float m2 = sh[0]; __syncthreads();
    float e2 = (j < NS_) ? __expf(p - m2) : 0.0f;
    sh[j] = e2; __syncthreads();
    for (int o = 64; o > 0; o >>= 1) { if (j < o) sh[j] += sh[j + o]; __syncthreads(); }
    float z = sh[0];
    int lab = slab2[b * T_ + t];
    if (j == lab) spick = p;
    __syncthreads();
    if (j == 0) atomicAdd(loss_acc, (m2 + logf(z) - spick) * (1.0f / NTOK));
}

__global__ void write_losses(const float* __restrict__ loss_acc, float* __restrict__ out) {
    out[OUT_ILOSS] = loss_acc[0];
    out[OUT_SLOSS] = loss_acc[1];
    out[OUT_TLOSS] = loss_acc[0] + loss_acc[1];
}

static inline dim3 gemm_grid(int M, int N) {
    int mt = (M + 15) / 16, nt = (N + 15) / 16;
    return dim3((mt + 1) / 2, (nt + 1) / 2);
}

// ------------------------------------------------------------------
extern "C" void kernel_launch(void* const* d_in, const int* in_sizes, int n_in,
                              void* d_out, int out_size, void* d_ws, size_t ws_size,
                              hipStream_t stream) {
    const float* lm1       = (const float*)d_in[0];
    const float* lm2       = (const float*)d_in[1];
    const int*   il1       = (const int*)d_in[2];
    const int*   il2       = (const int*)d_in[3];
    const int*   sl1       = (const int*)d_in[4];
    const int*   sl2       = (const int*)d_in[5];
    const float* iproto_in = (const float*)d_in[6];
    const int*   icnt_in   = (const int*)d_in[7];
    const float* sproto_in = (const float*)d_in[8];
    const int*   scnt_in   = (const int*)d_in[9];
    const float* W         = (const float*)d_in[10];   // (300,768) row-major
    const float* b_enc     = (const float*)d_in[11];
    const float* bO        = (const float*)d_in[12];
    const float* bX        = (const float*)d_in[13];

    float* w = (float*)d_ws;
    float* o = (float*)d_out;

    // 0) zero segment-sum + loss accumulators (graph-replay-safe)
    zero_kernel<<<(WS_ZERO_N + 255) / 256, 256, 0, stream>>>(w + WS_ISUM, WS_ZERO_N);

    // 1) encoder GEMMs: enc = lm @ W^T + b_enc   (M x 768) x (768 x 300)
    wmma_gemm_abt<<<gemm_grid(B_ * T_, E_), 32, 0, stream>>>(lm1, H_, W, H_, w + WS_ENC1, E_,
                                                             B_ * T_, E_, H_, b_enc);
    wmma_gemm_abt<<<gemm_grid(B_ * T_, E_), 32, 0, stream>>>(lm2, H_, W, H_, w + WS_ENC2, E_,
                                                             B_ * T_, E_, H_, b_enc);
    wmma_gemm_abt<<<gemm_grid(NS_, E_), 32, 0, stream>>>(sproto_in, H_, W, H_, w + WS_SPE, E_,
                                                         NS_, E_, H_, b_enc);

    // 2) prototype segment sums from enc1
    accum_intent<<<B_, 128, 0, stream>>>(w + WS_ENC1, il1, w + WS_ISUM, w + WS_ICNT);
    accum_slot<<<NTOK, 128, 0, stream>>>(w + WS_ENC1, sl1, w + WS_SSUM, w + WS_SCNT);

    // 3) finalize prototypes (-> d_out and workspace)
    finalize_intent<<<(NI_ * E_ + 255) / 256, 256, 0, stream>>>(
        iproto_in, icnt_in, w + WS_ISUM, w + WS_ICNT,
        o + OUT_IPROTO, w + WS_IPROTO, o + OUT_ICNT);
    finalize_slot<<<(NS_ * E_ + 255) / 256, 256, 0, stream>>>(
        w + WS_SPE, scnt_in, w + WS_SSUM, w + WS_SCNT,
        o + OUT_SPROTO, w + WS_SPROTO, o + OUT_SCNT);

    // 4) compact enc2 tokens 1..T-1 into dense buffer (reuses enc1 space)
    compact_toks<<<(NTOK * E_ + 255) / 256, 256, 0, stream>>>(w + WS_ENC2, w + WS_ENC1);

    // 5) norms for cosine similarity
    row_l2norm<<<NI_, 128, 0, stream>>>(w + WS_IPROTO, E_, E_, w + WS_IPNORM);
    row_l2norm<<<NS_, 128, 0, stream>>>(w + WS_SPROTO, E_, E_, w + WS_SPNORM);
    row_l2norm<<<B_, 128, 0, stream>>>(w + WS_ENC2, T_ * E_, E_, w + WS_CLSNORM);
    row_l2norm<<<NTOK, 128, 0, stream>>>(w + WS_ENC1, E_, E_, w + WS_TOKNORM);

    // 6) similarity GEMMs (raw dots; norms applied in softmax kernels)
    //    intent: cls2 rows are every T-th row of enc2 -> lda = T*E
    wmma_gemm_abt<<<gemm_grid(B_, NI_), 32, 0, stream>>>(w + WS_ENC2, T_ * E_,
                                                         w + WS_IPROTO, E_,
                                                         o + OUT_ILOGITS, NI_,
                                                         B_, NI_, E_, nullptr);
    wmma_gemm_abt<<<gemm_grid(NTOK, NS_), 32, 0, stream>>>(w + WS_ENC1, E_,
                                                           w + WS_SPROTO, E_,
                                                           o + OUT_SLOGITS, NS_,
                                                           NTOK, NS_, E_, nullptr);

    // 7) softmax + CE-on-probs, losses
    intent_softmax_loss<<<B_, 64, 0, stream>>>(o + OUT_ILOGITS, w + WS_CLSNORM,
                                               w + WS_IPNORM, il2, w + WS_LOSS);
    slot_softmax_loss<<<NTOK, 128, 0, stream>>>(o + OUT_SLOGITS, w + WS_TOKNORM,
                                                w + WS_SPNORM, sl2, bO, bX,
                                                w + WS_LOSS + 1);
    write_losses<<<1, 1, 0, stream>>>(w + WS_LOSS, o);
}